// SigLoss_24137716204245
// MI455X (gfx1250) — compile-verified
//
#include <hip/hip_runtime.h>
#include <hip/hip_bf16.h>

// SigLoss on MI455X (gfx1250):
//  kernel 1: sig_gram_wmma      — fp32 WMMA (16x16x4) Gram matrices, stores (M-1)
//  kernel 2: sig_pde_wavefront  — whole slice staged in 320KB WGP LDS via async
//                                 global->LDS b128, anti-diagonal Goursat scan
//  kernel 3: sig_reduce         — mean_b(kxx + kyy - 2 kxy)

typedef __attribute__((ext_vector_type(2))) float v2f;
typedef __attribute__((ext_vector_type(8))) float v8f;

namespace {
constexpr int kL      = 256;             // sequence length
constexpr int kD      = 64;              // feature dim
constexpr int kB      = 128;             // batch
constexpr int kM      = 255;             // M rows/cols = L-1
constexpr int kPitch  = 260;             // row pitch (words): 259 = 3 mod 64 -> conflict-free diagonals
constexpr int kSliceF = kM * kPitch;     // 66300 floats per (M-1) slice
constexpr int kSlices = 3 * kB;          // 384 (XX, YY, XY) x batch
constexpr int kSteps  = 2 * kM - 1;      // 509 anti-diagonal wavefront steps
}

// ---------------------------------------------------------------------------
// Kernel 1: Gram matrices via V_WMMA_F32_16X16X4_F32.
// Grid: 32 blocks per slice, 256 threads (8 waves), 1 wave per 16x16 tile.
// Accumulator starts at -1 so D = dx*dy^T + C == (M - 1) directly.
// ---------------------------------------------------------------------------
__global__ __launch_bounds__(256) void sig_gram_wmma(
    const float* __restrict__ X, const float* __restrict__ Y,
    float* __restrict__ mm1, int slice0, int nslices)
{
  const int lane       = threadIdx.x & 31;
  const int wave       = threadIdx.x >> 5;
  const int sliceLocal = blockIdx.x >> 5;            // uniform per block
  if (sliceLocal >= nslices) return;                 // whole-wave exit, EXEC stays full
  const int slice = slice0 + sliceLocal;
  const int pair  = slice >> 7;                      // 0: k(X,X)  1: k(Y,Y)  2: k(X,Y)
  const int b     = slice & 127;
  const int tile  = (blockIdx.x & 31) * 8 + wave;    // 256 tiles of 16x16 per slice
  const int ti    = tile >> 4;
  const int tj    = tile & 15;

  const float* __restrict__ Pa = (pair == 1) ? Y : X;
  const float* __restrict__ Pb = (pair == 0) ? X : Y;
  const float* Ap = Pa + (size_t)b * kL * kD;
  const float* Bp = Pb + (size_t)b * kL * kD;

  const int  hi   = lane >> 4;                       // 0: lanes 0-15, 1: lanes 16-31
  const int  l    = lane & 15;
  const int  rowA = ti * 16 + l;                     // dx row (M row)
  const int  rowB = tj * 16 + l;                     // dy row (M col)
  const bool va   = rowA < kM;
  const bool vb   = rowB < kM;
  const float* a0 = Ap + (size_t)(va ? rowA : 0) * kD;
  const float* b0 = Bp + (size_t)(vb ? rowB : 0) * kD;

  v8f acc = {-1.f, -1.f, -1.f, -1.f, -1.f, -1.f, -1.f, -1.f};
  const v2f zz = {0.f, 0.f};

#pragma unroll
  for (int c = 0; c < kD / 4; ++c) {
    // A 16x4 layout: lo half holds K = k,k+1 ; hi half K = k+2,k+3. B analogous.
    const int k = 4 * c + 2 * hi;
    v2f ax0 = *(const v2f*)(a0 + k);
    v2f ax1 = *(const v2f*)(a0 + kD + k);
    v2f bx0 = *(const v2f*)(b0 + k);
    v2f bx1 = *(const v2f*)(b0 + kD + k);
    v2f Aop = va ? (ax1 - ax0) : zz;                 // dx computed on the fly
    v2f Bop = vb ? (bx1 - bx0) : zz;                 // dy computed on the fly
    acc = __builtin_amdgcn_wmma_f32_16x16x4_f32(
        false, Aop, false, Bop, (short)0, acc, false, false);
  }

  float* out = mm1 + (size_t)sliceLocal * kSliceF;
#pragma unroll
  for (int r = 0; r < 8; ++r) {
    const int row = ti * 16 + r + 8 * hi;            // C/D layout: vgpr r -> M=r (+8 hi half)
    const int col = tj * 16 + l;
    if (row < kM && col < kM) out[row * kPitch + col] = acc[r];
  }
}

// ---------------------------------------------------------------------------
// Kernel 2: Goursat PDE wavefront. One block (8 waves) per slice.
// Whole (M-1) slice staged into dynamic LDS (265KB + 3KB diagonals of the
// 320KB WGP LDS) with gfx1250 async global->LDS b128 copies, then 509
// barriered anti-diagonal steps. Diagonal LDS reads are bank-conflict-free
// (pitch 260 -> word addr = 259*i + t, 259 coprime-mod-64 stride per lane).
// ---------------------------------------------------------------------------
__global__ __launch_bounds__(256) void sig_pde_wavefront(
    const float* __restrict__ mm1, float* __restrict__ partial, int slice0)
{
  extern __shared__ float lds[];                     // kSliceF + 3*256 floats
  float* Mld  = lds;
  float* Dbuf = lds + kSliceF;                       // 3 rotating K-diagonal buffers

  const int sliceLocal = blockIdx.x;
  const int slice      = slice0 + sliceLocal;
  const float* g = mm1 + (size_t)sliceLocal * kSliceF;

  // Async bulk copy: 16575 x 16B, per-lane LDS destination = truncated flat addr.
  const int nVec = kSliceF / 4;
  for (int v = threadIdx.x; v < nVec; v += blockDim.x) {
    unsigned long long ga = (unsigned long long)(const void*)(g + 4 * v);
    unsigned la = (unsigned)(unsigned long long)(const void*)(Mld + 4 * v);
    asm volatile("global_load_async_to_lds_b128 %0, %1, off"
                 :: "v"(la), "v"(ga) : "memory");
  }
  asm volatile("s_wait_asynccnt 0" ::: "memory");
  __syncthreads();

  const int i = threadIdx.x;                         // thread = M row
  for (int t = 0; t < kSteps; ++t) {
    float* At  = Dbuf + 256 * (t % 3);               // diagonal t   (write)
    float* At1 = Dbuf + 256 * ((t + 2) % 3);         // diagonal t-1 (read)
    float* At2 = Dbuf + 256 * ((t + 1) % 3);         // diagonal t-2 (read)
    const int j = t - i;
    if (i < kM && j >= 0 && j < kM) {
      // K[i+1][j+1] = K[i+1][j] + K[i][j+1] + K[i][j] * (M[i][j] - 1)
      float Lv = (j == 0) ? 1.f : At1[i];
      float Uv = (i == 0) ? 1.f : At1[i - 1];
      float Dv = (i == 0 || j == 0) ? 1.f : At2[i - 1];
      At[i] = Lv + Uv + Dv * Mld[i * kPitch + j];
    }
    __syncthreads();
  }
  if (i == kM - 1) partial[slice] = Dbuf[256 * ((kSteps - 1) % 3) + (kM - 1)];
}

// ---------------------------------------------------------------------------
// Kernel 3: loss = mean_b( kxx + kyy - 2*kxy )
// ---------------------------------------------------------------------------
__global__ __launch_bounds__(128) void sig_reduce(
    const float* __restrict__ partial, float* __restrict__ out)
{
  __shared__ float sh[128];
  const int b = threadIdx.x;
  sh[b] = partial[b] + partial[kB + b] - 2.f * partial[2 * kB + b];
  __syncthreads();
  for (int s = 64; s > 0; s >>= 1) {
    if (b < s) sh[b] += sh[b + s];
    __syncthreads();
  }
  if (b == 0) out[0] = sh[0] * (1.f / (float)kB);
}

extern "C" void kernel_launch(void* const* d_in, const int* in_sizes, int n_in,
                              void* d_out, int out_size, void* d_ws, size_t ws_size,
                              hipStream_t stream) {
  (void)in_sizes; (void)n_in; (void)out_size;
  const float* X = (const float*)d_in[0];
  const float* Y = (const float*)d_in[1];
  float* out     = (float*)d_out;
  float* partial = (float*)d_ws;                     // first 384 floats of ws
  float* mm1     = partial + 1024;                   // 4KB in, keeps 16B alignment

  const size_t sliceBytes = (size_t)kSliceF * sizeof(float);
  size_t avail = (ws_size > 4096) ? (ws_size - 4096) : 0;
  int cap = (int)(avail / sliceBytes);               // slices per chunk that fit in ws
  if (cap < 1) cap = 1;
  if (cap > kSlices) cap = kSlices;
  const size_t ldsBytes = (size_t)(kSliceF + 3 * 256) * sizeof(float); // ~268 KB

  for (int s0 = 0; s0 < kSlices; s0 += cap) {
    int ns = kSlices - s0;
    if (ns > cap) ns = cap;
    sig_gram_wmma<<<dim3((unsigned)(ns * 32)), dim3(256), 0, stream>>>(
        X, Y, mm1, s0, ns);
    sig_pde_wavefront<<<dim3((unsigned)ns), dim3(256), ldsBytes, stream>>>(
        mm1, partial, s0);
  }
  sig_reduce<<<dim3(1), dim3(128), 0, stream>>>(partial, out);
}